// MultiHeadAttention_25598005084251
// MI455X (gfx1250) — compile-verified
//
#include <hip/hip_runtime.h>
#include <hip/hip_bf16.h>

// MI455X (gfx1250) fused MHA, fp32 end-to-end via V_WMMA_F32_16X16X4_F32,
// with CDNA5 async global->LDS copies (ASYNCcnt) double-buffering the GEMM
// K-loops. Problem: B=4, S=2048, DM=512, H=8 (depth 64).
//
// Bandwidth-bound on materializing the 32x2048x2048 weights matrix (~2.1 GB
// of HBM traffic ~= 92us at 23.3 TB/s), so exact fp32 WMMA is the right
// precision: bf16 conversion would save no bandwidth and lose accuracy.
//
// d_out layout: [ out (4*2048*512) | weights (32*2048*2048) ]
// d_ws  layout: [ q | k | v (each 32*2048*64) | merged (4*2048*512) ] = 64 MB

typedef float v2f __attribute__((ext_vector_type(2)));
typedef float v8f __attribute__((ext_vector_type(8)));

#define S_LEN 2048
#define DMODEL 512
#define NBATCH 4
#define NHEAD 8
#define DEPTH 64

// ---- CDNA5 async global->LDS copy (tracked with ASYNCcnt) -----------------
// 16 bytes per lane, LDS dest offset in a VGPR, 64-bit global addr in a VGPR
// pair. Generic shared pointers carry the LDS offset in their low 32 bits.
__device__ __forceinline__ void async_copy16(const float* g, const float* l) {
  const uint32_t lds_off = (uint32_t)(uintptr_t)l;
  asm volatile("global_load_async_to_lds_b128 %0, %1, off"
               :
               : "v"(lds_off), "v"(g)
               : "memory");
}
__device__ __forceinline__ void wait_async0() {
  asm volatile("s_wait_asynccnt 0x0" ::: "memory");
}

// ---- WMMA fp32 fragment helpers (wave32, 16x16x4) -------------------------
// A 16x4 (MxK): lane L(0-15) m=L, k=k0+{0,1}; lanes 16-31 same m, k=k0+{2,3}.
__device__ __forceinline__ v2f lds_a_frag(const float* s, int ld, int mbase,
                                          int k0, int lane) {
  const float* p = s + (mbase + (lane & 15)) * ld + k0 + ((lane >> 4) << 1);
  return v2f{p[0], p[1]};
}
// B 4x16 (KxN): lane n(0-15) v0=B[k0][n], v1=B[k0+1][n]; lanes 16-31 rows +2.
__device__ __forceinline__ v2f lds_b_frag(const float* s, int ld, int k0,
                                          int nbase, int lane) {
  const float* p = s + (k0 + ((lane >> 4) << 1)) * ld + nbase + (lane & 15);
  return v2f{p[0], p[ld]};
}

__device__ __forceinline__ v8f wmma_f32(v2f a, v2f b, v8f c) {
  return __builtin_amdgcn_wmma_f32_16x16x4_f32(false, a, false, b, (short)0, c,
                                               false, false);
}

// LDS pitches (floats). Chosen so fragment reads are bank-conflict-free and
// every row start is 16B-aligned for b128 async stores:
//   XS_LD=36: {36*m mod 64} hits each multiple of 4 once -> A b64 reads clean
//   WS_LD=72: 2*72 mod 64 = 16 -> upper/lower half-wave B reads disjoint
//   QS_LD=68: {4m+k+j} covers all 64 banks exactly once
#define XS_LD 36
#define WS_LD 72
#define QS_LD 68
#define KT_LD 72

// ---- GEMM: out = X(MxK) @ W(KxN) + bias; optional head-split epilogue -----
// Block: 128 threads (4 waves). Tile 64(M) x 64(N), K chunk 32, async
// double-buffered LDS staging. Wave w: rows [w*16,w*16+16) x 64 cols.
__device__ __forceinline__ void gemm_issue_tile(const float* __restrict__ X,
                                                const float* __restrict__ W,
                                                int K, int N, int m0, int n0,
                                                int k0, float* xbuf,
                                                float* wbuf, int tid) {
#pragma unroll
  for (int i = 0; i < 4; ++i) {  // Xs: 64x32 floats = 512 float4
    int idx = tid + i * 128;
    int r = idx >> 3, c4 = (idx & 7) << 2;
    async_copy16(X + (size_t)(m0 + r) * K + k0 + c4, xbuf + r * XS_LD + c4);
  }
#pragma unroll
  for (int i = 0; i < 4; ++i) {  // Ws: 32x64 floats = 512 float4
    int idx = tid + i * 128;
    int r = idx >> 4, c4 = (idx & 15) << 2;
    async_copy16(W + (size_t)(k0 + r) * N + n0 + c4, wbuf + r * WS_LD + c4);
  }
}

__global__ __launch_bounds__(128) void gemm_bias_kernel(
    const float* __restrict__ X, const float* __restrict__ W,
    const float* __restrict__ bias, float* __restrict__ out, int M, int K,
    int N, int split_heads) {
  __shared__ float Xs[2][64 * XS_LD];
  __shared__ float Ws[2][32 * WS_LD];
  const int tid = threadIdx.x;
  const int lane = tid & 31, wave = tid >> 5;
  const int m0 = blockIdx.y * 64, n0 = blockIdx.x * 64;

  v8f c[4] = {};

  gemm_issue_tile(X, W, K, N, m0, n0, 0, Xs[0], Ws[0], tid);
  wait_async0();
  __syncthreads();

  int buf = 0;
  for (int k0 = 0; k0 < K; k0 += 32) {
    const float* xb = Xs[buf];
    const float* wb = Ws[buf];
    if (k0 + 32 < K)
      gemm_issue_tile(X, W, K, N, m0, n0, k0 + 32, Xs[buf ^ 1], Ws[buf ^ 1],
                      tid);
#pragma unroll
    for (int kk = 0; kk < 32; kk += 4) {
      v2f a = lds_a_frag(xb, XS_LD, wave * 16, kk, lane);
#pragma unroll
      for (int t = 0; t < 4; ++t) {
        v2f b = lds_b_frag(wb, WS_LD, kk, t * 16, lane);
        c[t] = wmma_f32(a, b, c[t]);
      }
    }
    wait_async0();
    __syncthreads();
    buf ^= 1;
  }

  // Epilogue. C layout: VGPR r -> m = base + r (+8 for lanes 16-31), n=lane&15
  const int mbase = m0 + wave * 16 + ((lane >> 4) << 3);
#pragma unroll
  for (int t = 0; t < 4; ++t) {
    const int n = n0 + t * 16 + (lane & 15);
    const float bv = bias[n];
#pragma unroll
    for (int r = 0; r < 8; ++r) {
      const int m = mbase + r;
      const float val = c[t][r] + bv;
      if (split_heads) {
        // m = b*S + s ; n = h*depth + d  ->  [(h*B+b), s, d]
        const int bb = m >> 11, ss = m & (S_LEN - 1);
        const int hh = n >> 6, dd = n & (DEPTH - 1);
        out[((size_t)(hh * NBATCH + bb) * S_LEN + ss) * DEPTH + dd] = val;
      } else {
        out[(size_t)m * N + n] = val;
      }
    }
  }
}

// ---- scores = q @ k^T / 8, per head-batch --------------------------------
__global__ __launch_bounds__(128) void scores_kernel(
    const float* __restrict__ q, const float* __restrict__ k,
    float* __restrict__ scores) {
  __shared__ float Qs[64 * QS_LD];
  __shared__ float KT[64 * KT_LD];  // transposed: KT[d][j]
  const int tid = threadIdx.x, lane = tid & 31, wave = tid >> 5;
  const int hb = blockIdx.z;
  const int i0 = blockIdx.y * 64, j0 = blockIdx.x * 64;
  const float* qb = q + (size_t)hb * S_LEN * DEPTH;
  const float* kb = k + (size_t)hb * S_LEN * DEPTH;

  // Qs: async copy 64x64 tile straight into LDS (1024 x b128, 8 per thread)
#pragma unroll
  for (int i = 0; i < 8; ++i) {
    int idx = tid + i * 128;
    int r = idx >> 4, c4 = (idx & 15) << 2;
    async_copy16(qb + (size_t)(i0 + r) * DEPTH + c4, Qs + r * QS_LD + c4);
  }
  // K tile must be transposed -> stage through VGPRs
#pragma unroll
  for (int i = 0; i < 8; ++i) {
    int idx = tid + i * 128;
    int j = idx >> 4, c4 = (idx & 15) << 2;
    float4 v = *(const float4*)(kb + (size_t)(j0 + j) * DEPTH + c4);
    KT[(c4 + 0) * KT_LD + j] = v.x;
    KT[(c4 + 1) * KT_LD + j] = v.y;
    KT[(c4 + 2) * KT_LD + j] = v.z;
    KT[(c4 + 3) * KT_LD + j] = v.w;
  }
  wait_async0();
  __syncthreads();

  v8f c[4] = {};
#pragma unroll
  for (int kk = 0; kk < DEPTH; kk += 4) {
    v2f a = lds_a_frag(Qs, QS_LD, wave * 16, kk, lane);
#pragma unroll
    for (int t = 0; t < 4; ++t) {
      v2f b = lds_b_frag(KT, KT_LD, kk, t * 16, lane);
      c[t] = wmma_f32(a, b, c[t]);
    }
  }

  const float scale = 0.125f;  // 1/sqrt(64)
  const int mbase = i0 + wave * 16 + ((lane >> 4) << 3);
  float* sb = scores + (size_t)hb * S_LEN * S_LEN;
#pragma unroll
  for (int t = 0; t < 4; ++t) {
    const int n = j0 + t * 16 + (lane & 15);
#pragma unroll
    for (int r = 0; r < 8; ++r) {
      sb[(size_t)(mbase + r) * S_LEN + n] = c[t][r] * scale;
    }
  }
}

// ---- row softmax over weights (in place), applying mask*(-1e9) -----------
__global__ __launch_bounds__(256) void softmax_kernel(
    float* __restrict__ w, const float* __restrict__ mask) {
  const int row = blockIdx.x;  // hb*S + i
  const int i = row & (S_LEN - 1);
  float* rp = w + (size_t)row * S_LEN;
  const float* mp = mask + (size_t)i * S_LEN;
  const int lane = threadIdx.x & 31, wv = threadIdx.x >> 5;
  __shared__ float red[8];

  float vals[8];
  float mx = -3.4e38f;
#pragma unroll
  for (int t = 0; t < 8; ++t) {
    const int cix = threadIdx.x + t * 256;
    const float v = rp[cix] + mp[cix] * (-1e9f);
    vals[t] = v;
    mx = fmaxf(mx, v);
  }
#pragma unroll
  for (int off = 16; off > 0; off >>= 1)
    mx = fmaxf(mx, __shfl_xor(mx, off, 32));
  if (lane == 0) red[wv] = mx;
  __syncthreads();
  mx = red[0];
#pragma unroll
  for (int t = 1; t < 8; ++t) mx = fmaxf(mx, red[t]);
  __syncthreads();

  float sum = 0.f;
#pragma unroll
  for (int t = 0; t < 8; ++t) {
    vals[t] = __expf(vals[t] - mx);
    sum += vals[t];
  }
#pragma unroll
  for (int off = 16; off > 0; off >>= 1) sum += __shfl_xor(sum, off, 32);
  if (lane == 0) red[wv] = sum;
  __syncthreads();
  sum = 0.f;
#pragma unroll
  for (int t = 0; t < 8; ++t) sum += red[t];
  const float inv = 1.0f / sum;
#pragma unroll
  for (int t = 0; t < 8; ++t) {
    const int cix = threadIdx.x + t * 256;
    rp[cix] = vals[t] * inv;
  }
}

// ---- attn = weights @ v, fused head-merge epilogue ------------------------
__device__ __forceinline__ void attn_issue_tile(const float* __restrict__ wb,
                                                const float* __restrict__ vb,
                                                int i0, int k0, float* abuf,
                                                float* vbuf, int tid) {
#pragma unroll
  for (int i = 0; i < 4; ++i) {  // As: 64x32
    int idx = tid + i * 128;
    int r = idx >> 3, c4 = (idx & 7) << 2;
    async_copy16(wb + (size_t)(i0 + r) * S_LEN + k0 + c4,
                 abuf + r * XS_LD + c4);
  }
#pragma unroll
  for (int i = 0; i < 4; ++i) {  // Vs: 32x64
    int idx = tid + i * 128;
    int r = idx >> 4, c4 = (idx & 15) << 2;
    async_copy16(vb + (size_t)(k0 + r) * DEPTH + c4, vbuf + r * WS_LD + c4);
  }
}

__global__ __launch_bounds__(128) void attn_v_kernel(
    const float* __restrict__ wts, const float* __restrict__ v,
    float* __restrict__ merged) {
  __shared__ float As[2][64 * XS_LD];
  __shared__ float Vs[2][32 * WS_LD];
  const int tid = threadIdx.x, lane = tid & 31, wave = tid >> 5;
  const int hb = blockIdx.z;
  const int i0 = blockIdx.y * 64;
  const float* wb = wts + (size_t)hb * S_LEN * S_LEN;
  const float* vb = v + (size_t)hb * S_LEN * DEPTH;

  v8f c[4] = {};

  attn_issue_tile(wb, vb, i0, 0, As[0], Vs[0], tid);
  wait_async0();
  __syncthreads();

  int buf = 0;
  for (int k0 = 0; k0 < S_LEN; k0 += 32) {
    const float* ab = As[buf];
    const float* vv = Vs[buf];
    if (k0 + 32 < S_LEN)
      attn_issue_tile(wb, vb, i0, k0 + 32, As[buf ^ 1], Vs[buf ^ 1], tid);
#pragma unroll
    for (int kk = 0; kk < 32; kk += 4) {
      v2f a = lds_a_frag(ab, XS_LD, wave * 16, kk, lane);
#pragma unroll
      for (int t = 0; t < 4; ++t) {
        v2f b = lds_b_frag(vv, WS_LD, kk, t * 16, lane);
        c[t] = wmma_f32(a, b, c[t]);
      }
    }
    wait_async0();
    __syncthreads();
    buf ^= 1;
  }

  // hb = h*B + b ; write merged[(b*S + s)*DM + h*depth + n]
  const int hh = hb >> 2, bb = hb & 3;
  const int mbase = i0 + wave * 16 + ((lane >> 4) << 3);
#pragma unroll
  for (int t = 0; t < 4; ++t) {
    const int n = t * 16 + (lane & 15);
#pragma unroll
    for (int r = 0; r < 8; ++r) {
      const int s = mbase + r;
      merged[(size_t)(bb * S_LEN + s) * DMODEL + hh * DEPTH + n] = c[t][r];
    }
  }
}

extern "C" void kernel_launch(void* const* d_in, const int* in_sizes, int n_in,
                              void* d_out, int out_size, void* d_ws,
                              size_t ws_size, hipStream_t stream) {
  (void)in_sizes; (void)n_in; (void)out_size; (void)ws_size;
  const float* Q = (const float*)d_in[0];
  const float* Kin = (const float*)d_in[1];
  const float* V = (const float*)d_in[2];
  const float* mask = (const float*)d_in[3];
  const float* Wq_w = (const float*)d_in[4];
  const float* Wq_b = (const float*)d_in[5];
  const float* Wk_w = (const float*)d_in[6];
  const float* Wk_b = (const float*)d_in[7];
  const float* Wv_w = (const float*)d_in[8];
  const float* Wv_b = (const float*)d_in[9];
  const float* Wo_w = (const float*)d_in[10];
  const float* Wo_b = (const float*)d_in[11];

  float* out = (float*)d_out;                              // (4,2048,512)
  float* weights = out + (size_t)NBATCH * S_LEN * DMODEL;  // (32,2048,2048)

  float* qs = (float*)d_ws;
  float* ks = qs + (size_t)NHEAD * NBATCH * S_LEN * DEPTH;
  float* vs = ks + (size_t)NHEAD * NBATCH * S_LEN * DEPTH;
  float* merged = vs + (size_t)NHEAD * NBATCH * S_LEN * DEPTH;

  const dim3 thr128(128);
  const dim3 gGemm(DMODEL / 64, (NBATCH * S_LEN) / 64);  // (8, 128)

  gemm_bias_kernel<<<gGemm, thr128, 0, stream>>>(
      Q, Wq_w, Wq_b, qs, NBATCH * S_LEN, DMODEL, DMODEL, 1);
  gemm_bias_kernel<<<gGemm, thr128, 0, stream>>>(
      Kin, Wk_w, Wk_b, ks, NBATCH * S_LEN, DMODEL, DMODEL, 1);
  gemm_bias_kernel<<<gGemm, thr128, 0, stream>>>(
      V, Wv_w, Wv_b, vs, NBATCH * S_LEN, DMODEL, DMODEL, 1);

  scores_kernel<<<dim3(S_LEN / 64, S_LEN / 64, NHEAD * NBATCH), thr128, 0,
                  stream>>>(qs, ks, weights);

  softmax_kernel<<<dim3(NHEAD * NBATCH * S_LEN), dim3(256), 0, stream>>>(
      weights, mask);

  attn_v_kernel<<<dim3(1, S_LEN / 64, NHEAD * NBATCH), thr128, 0, stream>>>(
      weights, vs, merged);

  gemm_bias_kernel<<<gGemm, thr128, 0, stream>>>(
      merged, Wo_w, Wo_b, out, NBATCH * S_LEN, DMODEL, DMODEL, 0);
}